// CoherenceLoss_89352499626065
// MI455X (gfx1250) — compile-verified
//
#include <hip/hip_runtime.h>
#include <math.h>

typedef float v2f __attribute__((ext_vector_type(2)));
typedef float v8f __attribute__((ext_vector_type(8)));

#define NTOPIC 100
#define VOCAB  8192
#define TOPK   20

// ---- kernel 2 tiling ----
#define BTOP   16                 // topics per block
#define SLOTS  (BTOP * TOPK)      // 320 K-slots (block-diagonal A columns)
#define KCH    80                 // K-chunk staged in LDS at a time
#define CCOLS  128                // column chunk (8 waves * 16 cols)
#define BSTR   136                // padded LDS row stride (words) -> conflict-free

// --------------------------------------------------------------------------
__global__ void k0_zero(int* __restrict__ colCnt) {
    int i = blockIdx.x * blockDim.x + threadIdx.x;
    if (i < VOCAB) colCnt[i] = 0;
}

// --------------------------------------------------------------------------
// Per topic row: full max + sum(exp) for softmax(beta), top-20 selection,
// softmax weights over the top-20 (== softmax(beta + Mc_add) exactly, since
// exp(x-99999-max) underflows to 0 in fp32), and column counts for Md.
__global__ __launch_bounds__(256) void k1_topk(const float* __restrict__ beta,
                                               int*   __restrict__ topIdx,
                                               float* __restrict__ topW,
                                               float* __restrict__ rowMax,
                                               float* __restrict__ rowSum,
                                               int*   __restrict__ colCnt) {
    __shared__ float sRow[VOCAB];
    __shared__ float sVal[256];
    __shared__ int   sIdx[256];
    __shared__ float sTopVal[TOPK];
    __shared__ int   sTopIdx[TOPK];
    __shared__ float sScal;

    const int k   = blockIdx.x;
    const int tid = threadIdx.x;
    const float* row = beta + (size_t)k * VOCAB;

    for (int i = tid; i < VOCAB; i += 256) sRow[i] = row[i];
    __syncthreads();

    // full-row max
    float m = -INFINITY;
    for (int i = tid; i < VOCAB; i += 256) m = fmaxf(m, sRow[i]);
    sVal[tid] = m; __syncthreads();
    for (int o = 128; o > 0; o >>= 1) {
        if (tid < o) sVal[tid] = fmaxf(sVal[tid], sVal[tid + o]);
        __syncthreads();
    }
    const float fullMax = sVal[0];
    __syncthreads();

    // sum(exp(x - max)) over full row
    float se = 0.f;
    for (int i = tid; i < VOCAB; i += 256) se += expf(sRow[i] - fullMax);
    sVal[tid] = se; __syncthreads();
    for (int o = 128; o > 0; o >>= 1) {
        if (tid < o) sVal[tid] += sVal[tid + o];
        __syncthreads();
    }
    const float sumExp = sVal[0];
    __syncthreads();

    // top-20: iterated argmax, smallest-index tie-break (first occurrence)
    for (int sel = 0; sel < TOPK; ++sel) {
        float bv = -INFINITY; int bi = 0x7fffffff;
        for (int i = tid; i < VOCAB; i += 256) {
            float v = sRow[i];
            if (v > bv || (v == bv && i < bi)) { bv = v; bi = i; }
        }
        sVal[tid] = bv; sIdx[tid] = bi; __syncthreads();
        for (int o = 128; o > 0; o >>= 1) {
            if (tid < o) {
                float v2 = sVal[tid + o]; int i2 = sIdx[tid + o];
                if (v2 > sVal[tid] || (v2 == sVal[tid] && i2 < sIdx[tid])) {
                    sVal[tid] = v2; sIdx[tid] = i2;
                }
            }
            __syncthreads();
        }
        if (tid == 0) {
            sTopVal[sel] = sVal[0]; sTopIdx[sel] = sIdx[0];
            sRow[sIdx[0]] = -INFINITY;
        }
        __syncthreads();
    }

    if (tid == 0) {
        float mm = sTopVal[0], s = 0.f;
        for (int i = 0; i < TOPK; ++i) s += expf(sTopVal[i] - mm);
        sScal = s;
        rowMax[k] = fullMax;
        rowSum[k] = sumExp;
    }
    __syncthreads();
    if (tid < TOPK) {
        float w = expf(sTopVal[tid] - sTopVal[0]) / sScal;
        topIdx[k * TOPK + tid] = sTopIdx[tid];
        topW [k * TOPK + tid] = w;
        atomicAdd(&colCnt[sTopIdx[tid]], 1);   // integer atomic: deterministic
    }
}

// --------------------------------------------------------------------------
// M = P @ W as a block-dense GEMM per 16-topic block.
// A[16 x 320] block-diagonal (col s nonzero only at row s/20, value = w),
// B[320 x 8192] = gathered W rows, staged through LDS 80x128 fp32 tiles.
// Each wave: v_wmma_f32_16x16x4_f32 over K (two independent C chains for
// ILP), one 16x16 C tile per column tile.
__global__ __launch_bounds__(256) void k2_wmma(const float* __restrict__ Wmat,
                                               const int*   __restrict__ topIdx,
                                               const float* __restrict__ topW,
                                               float*       __restrict__ Mout) {
    __shared__ alignas(16) float Blds[KCH * BSTR];   // 80 * 136 * 4B = 42.5 KB
    __shared__ int               sI[SLOTS];
    __shared__ alignas(8)  float sW[SLOTS];

    const int tid       = threadIdx.x;
    const int topicBase = blockIdx.x * BTOP;
    const int nTopics   = (NTOPIC - topicBase < BTOP) ? (NTOPIC - topicBase) : BTOP;
    const int nSlots    = nTopics * TOPK;

    for (int s = tid; s < SLOTS; s += 256) {
        int t = s / TOPK;
        if (t < nTopics) {
            sI[s] = topIdx[(topicBase + t) * TOPK + (s % TOPK)];
            sW[s] = topW [(topicBase + t) * TOPK + (s % TOPK)];
        } else { sI[s] = 0; sW[s] = 0.f; }
    }
    __syncthreads();

    const int lane    = tid & 31;
    const int wv      = tid >> 5;       // 8 waves
    const int hi      = lane >> 4;      // half-wave selector (K 0,1 vs 2,3)
    const int r       = lane & 15;      // A row / B,C column within tile
    const int waveOff = wv * 16;
    const int sLo     = r * TOPK;       // lane's nonzero A-column range start

    for (int chunk = 0; chunk < VOCAB / CCOLS; ++chunk) {
        const int colBase = chunk * CCOLS;
        union { v8f v; float f[8]; } acc0, acc1;
        #pragma unroll
        for (int j = 0; j < 8; ++j) { acc0.f[j] = 0.f; acc1.f[j] = 0.f; }

        for (int kc = 0; kc < SLOTS / KCH; ++kc) {
            if (kc * KCH >= nSlots) break;      // block-uniform: barrier-safe
            __syncthreads();
            // stage gathered B tile [KCH x CCOLS] with 16B vector ops
            for (int e = tid; e < KCH * (CCOLS / 4); e += 256) {
                int srow = e >> 5;              // CCOLS/4 == 32
                int c4   = e & 31;
                int s    = kc * KCH + srow;
                float4 val = make_float4(0.f, 0.f, 0.f, 0.f);
                if (s < nSlots)
                    val = *(const float4*)(Wmat + (size_t)sI[s] * VOCAB
                                                + colBase + c4 * 4);
                *(float4*)(&Blds[srow * BSTR + c4 * 4]) = val;
            }
            __syncthreads();

            #pragma unroll
            for (int kk = 0; kk < KCH / 4; ++kk) {
                // A fragment: lanes 0-15 hold K=0,1; lanes 16-31 hold K=2,3.
                // Unconditional ds_load_b64 + v_cndmask (no exec masking).
                int klx = kc * KCH + kk * 4 + hi * 2;   // even -> 8B aligned
                v2f aw = *(const v2f*)(sW + klx);
                v2f a, b;
                a.x = ((unsigned)(klx     - sLo) < TOPK) ? aw.x : 0.f;
                a.y = ((unsigned)(klx + 1 - sLo) < TOPK) ? aw.y : 0.f;
                // B fragment: v0 = rows K=0/2, v1 = rows K=1/3, cols = lane&15
                int brow = kk * 4 + hi * 2;
                b.x = Blds[ brow      * BSTR + waveOff + r];
                b.y = Blds[(brow + 1) * BSTR + waveOff + r];
                if (kk & 1)
                    acc1.v = __builtin_amdgcn_wmma_f32_16x16x4_f32(
                                false, a, false, b, (short)0, acc1.v, false, false);
                else
                    acc0.v = __builtin_amdgcn_wmma_f32_16x16x4_f32(
                                false, a, false, b, (short)0, acc0.v, false, false);
            }
        }
        // C layout: VGPR j -> M-row j (lanes 0-15) / j+8 (lanes 16-31)
        #pragma unroll
        for (int j = 0; j < 8; ++j) {
            int t = topicBase + (hi ? j + 8 : j);
            if (t < NTOPIC)
                Mout[(size_t)t * VOCAB + colBase + waveOff + r] =
                    acc0.f[j] + acc1.f[j];
        }
    }
}

// --------------------------------------------------------------------------
__global__ __launch_bounds__(256) void k2b_minmax(const float* __restrict__ M,
                                                  float* __restrict__ Mmin,
                                                  float* __restrict__ Mmax) {
    __shared__ float sMin[256], sMax[256];
    const int t = blockIdx.x, tid = threadIdx.x;
    const float* row = M + (size_t)t * VOCAB;
    float mn = INFINITY, mx = -INFINITY;
    for (int i = tid; i < VOCAB; i += 256) {
        float v = row[i];
        mn = fminf(mn, v); mx = fmaxf(mx, v);
    }
    sMin[tid] = mn; sMax[tid] = mx; __syncthreads();
    for (int o = 128; o > 0; o >>= 1) {
        if (tid < o) {
            sMin[tid] = fminf(sMin[tid], sMin[tid + o]);
            sMax[tid] = fmaxf(sMax[tid], sMax[tid + o]);
        }
        __syncthreads();
    }
    if (tid == 0) { Mmin[t] = sMin[0]; Mmax[t] = sMax[0]; }
}

// --------------------------------------------------------------------------
// loss = 100 * softmax(beta)^2 * (1 - norm(M)); weighted by Md mask, reduced
// per row (one WG per row -> deterministic).
__global__ __launch_bounds__(256) void k3_loss(const float* __restrict__ beta,
                                               const float* __restrict__ M,
                                               const float* __restrict__ rowMax,
                                               const float* __restrict__ rowSum,
                                               const float* __restrict__ Mmin,
                                               const float* __restrict__ Mmax,
                                               const int*   __restrict__ topIdx,
                                               const int*   __restrict__ colCnt,
                                               float*       __restrict__ rowLoss) {
    __shared__ int   sTop[TOPK];
    __shared__ float sRed[256];
    const int t = blockIdx.x, tid = threadIdx.x;
    if (tid < TOPK) sTop[tid] = topIdx[t * TOPK + tid];
    __syncthreads();

    const float rm  = rowMax[t];
    const float inv = 1.f / rowSum[t];
    const float mn  = Mmin[t];
    const float invRange = 1.f / (Mmax[t] - mn);

    float acc = 0.f;
    for (int v = tid; v < VOCAB; v += 256) {
        float q    = expf(beta[(size_t)t * VOCAB + v] - rm) * inv;
        float wc   = 1.f - (M[(size_t)t * VOCAB + v] - mn) * invRange;
        float loss = 100.f * q * q * wc;
        int isTop = 0;
        #pragma unroll
        for (int i = 0; i < TOPK; ++i) isTop |= (sTop[i] == v);
        int md = (colCnt[v] - isTop) > 0;
        acc += loss * (md ? 0.7f : 0.3f);   // lambda_d / (1-lambda_d)
    }
    sRed[tid] = acc; __syncthreads();
    for (int o = 128; o > 0; o >>= 1) {
        if (tid < o) sRed[tid] += sRed[tid + o];
        __syncthreads();
    }
    if (tid == 0) rowLoss[t] = sRed[0];
}

// --------------------------------------------------------------------------
__global__ void k4_final(const float* __restrict__ rowLoss,
                         const int*   __restrict__ epoch,
                         float*       __restrict__ out) {
    float s = 0.f;
    for (int i = 0; i < NTOPIC; ++i) s += rowLoss[i];   // fixed order
    int e = epoch[0];
    // warmup_epochs = 100, lambda_a_delta = 100/100 = 1.0
    float lam = (e < 100) ? (float)e : 100.0f;
    out[0] = lam * 2.0f * s;
}

// --------------------------------------------------------------------------
extern "C" void kernel_launch(void* const* d_in, const int* in_sizes, int n_in,
                              void* d_out, int out_size, void* d_ws, size_t ws_size,
                              hipStream_t stream) {
    (void)in_sizes; (void)n_in; (void)out_size; (void)ws_size;
    const float* beta  = (const float*)d_in[0];
    const float* Wmat  = (const float*)d_in[1];
    const int*   epoch = (const int*)d_in[2];

    char* ws = (char*)d_ws;
    size_t off = 0;
    float* M       = (float*)(ws + off); off += (size_t)NTOPIC * VOCAB * sizeof(float); // 3.125 MB
    int*   topIdx  = (int*)  (ws + off); off += 8192;                                   // 100*20 ints
    float* topW    = (float*)(ws + off); off += 8192;
    float* rowMax  = (float*)(ws + off); off += 512;
    float* rowSum  = (float*)(ws + off); off += 512;
    int*   colCnt  = (int*)  (ws + off); off += VOCAB * sizeof(int);
    float* Mmin    = (float*)(ws + off); off += 512;
    float* Mmax    = (float*)(ws + off); off += 512;
    float* rowLoss = (float*)(ws + off); off += 512;

    k0_zero   <<<(VOCAB + 255) / 256, 256, 0, stream>>>(colCnt);
    k1_topk   <<<NTOPIC, 256, 0, stream>>>(beta, topIdx, topW, rowMax, rowSum, colCnt);
    k2_wmma   <<<(NTOPIC + BTOP - 1) / BTOP, 256, 0, stream>>>(Wmat, topIdx, topW, M);
    k2b_minmax<<<NTOPIC, 256, 0, stream>>>(M, Mmin, Mmax);
    k3_loss   <<<NTOPIC, 256, 0, stream>>>(beta, M, rowMax, rowSum, Mmin, Mmax,
                                           topIdx, colCnt, rowLoss);
    k4_final  <<<1, 1, 0, stream>>>(rowLoss, epoch, (float*)d_out);
}